// UNetSphericalHealpixResidualLongConnections_10316511445499
// MI455X (gfx1250) — compile-verified
//
#include <hip/hip_runtime.h>
#include <hip/hip_bf16.h>
#include <stddef.h>

#define B_SZ 4
#define NV0 12288
#define NV1 3072
#define NV2 768
#define NNB 21
#define ROWS0 (B_SZ * NV0)
#define ROWS1 (B_SZ * NV1)
#define ROWS2 (B_SZ * NV2)

typedef float v2f __attribute__((ext_vector_type(2)));
typedef float v8f __attribute__((ext_vector_type(8)));

// ---------------------------------------------------------------------------
// Sparse Laplacian multiply:  out[b,v,:] = sum_n val[v,n] * in[b,idx[v,n],:]
// mode sub!=null:             out = 2 * (L in) - sub      (Chebyshev T2 term)
// One block per (v, b); neighbor lists staged in LDS; float4 channel loop.
// ---------------------------------------------------------------------------
__global__ void lmul_kernel(const float* __restrict__ xin,
                            const int* __restrict__ idx,
                            const float* __restrict__ val,
                            const float* __restrict__ sub,
                            float* __restrict__ out,
                            int V, int F)
{
    const int v = blockIdx.x;
    const int b = blockIdx.y;
    __shared__ int   s_idx[NNB];
    __shared__ float s_val[NNB];
    if (threadIdx.x < NNB) {
        s_idx[threadIdx.x] = idx[v * NNB + threadIdx.x];
        s_val[threadIdx.x] = val[v * NNB + threadIdx.x];
    }
    __syncthreads();

    const int C4 = F >> 2;                    // F is always a multiple of 4
    const float4* x4 = (const float4*)xin;
    const size_t baseRow = (size_t)b * V;

    for (int c = threadIdx.x; c < C4; c += blockDim.x) {
        float4 acc = make_float4(0.f, 0.f, 0.f, 0.f);
        #pragma unroll 3
        for (int n = 0; n < NNB; ++n) {
            const float w = s_val[n];
            const float4 t = x4[(baseRow + (size_t)s_idx[n]) * C4 + c];
            acc.x += w * t.x; acc.y += w * t.y;
            acc.z += w * t.z; acc.w += w * t.w;
        }
        const size_t o = (baseRow + v) * (size_t)C4 + c;
        if (sub) {
            const float4 s4 = ((const float4*)sub)[o];
            acc.x = 2.f * acc.x - s4.x; acc.y = 2.f * acc.y - s4.y;
            acc.z = 2.f * acc.z - s4.z; acc.w = 2.f * acc.w - s4.w;
        }
        ((float4*)out)[o] = acc;
    }
}

// ---------------------------------------------------------------------------
// WMMA fp32 GEMM over up to 3 A-segments (Chebyshev terms kept separate):
//   out[M,N] = [A0|A1|A2][M, nseg*F] @ W[nseg*F, N] + bias
// Wave32; each wave owns a 64x16 macro-tile = four 16x16 fp32 accumulators
// stacked along M, so every K-step reuses one B (weight) fragment for 4
// V_WMMA_F32_16X16X4_F32 issues (4 wmma / 6 loads).  Guarantees:
// M % 256 == 0 (4 waves x 64 rows / block), N % 16 == 0, F % 4 == 0.
// A fragment: lane<16 -> K={k,k+1}, lane>=16 -> K={k+2,k+3}, row = m+(lane&15)
// C/D tile  : vgpr r, row = m + r + 8*(lane>>4), col = n0 + (lane&15)
// ---------------------------------------------------------------------------
__global__ void gemm_wmma(const float* __restrict__ A0,
                          const float* __restrict__ A1,
                          const float* __restrict__ A2,
                          const float* __restrict__ W,
                          const float* __restrict__ bias,
                          float* __restrict__ out,
                          int M, int N, int F, int nseg)
{
    const int wave = threadIdx.x >> 5;
    const int lane = threadIdx.x & 31;
    const int half = lane >> 4;
    const int lr   = lane & 15;
    const int m0   = (blockIdx.x * 4 + wave) * 64;
    const int n0   = blockIdx.y * 16;

    v8f acc0 = {}, acc1 = {}, acc2 = {}, acc3 = {};
    for (int s = 0; s < nseg; ++s) {
        const float* A = (s == 0) ? A0 : (s == 1 ? A1 : A2);
        const float* Wseg = W + (size_t)s * F * N;
        const float* a0r = A + (size_t)(m0 + lr) * F;
        const float* a1r = a0r + (size_t)16 * F;
        const float* a2r = a0r + (size_t)32 * F;
        const float* a3r = a0r + (size_t)48 * F;
        for (int k = 0; k < F; k += 4) {
            const int ka = k + 2 * half;
            v2f b;  b.x  = Wseg[(size_t)ka * N + n0 + lr];
                    b.y  = Wseg[(size_t)(ka + 1) * N + n0 + lr];
            v2f a0; a0.x = a0r[ka]; a0.y = a0r[ka + 1];
            v2f a1; a1.x = a1r[ka]; a1.y = a1r[ka + 1];
            v2f a2; a2.x = a2r[ka]; a2.y = a2r[ka + 1];
            v2f a3; a3.x = a3r[ka]; a3.y = a3r[ka + 1];
            acc0 = __builtin_amdgcn_wmma_f32_16x16x4_f32(
                false, a0, false, b, (short)0, acc0, false, false);
            acc1 = __builtin_amdgcn_wmma_f32_16x16x4_f32(
                false, a1, false, b, (short)0, acc1, false, false);
            acc2 = __builtin_amdgcn_wmma_f32_16x16x4_f32(
                false, a2, false, b, (short)0, acc2, false, false);
            acc3 = __builtin_amdgcn_wmma_f32_16x16x4_f32(
                false, a3, false, b, (short)0, acc3, false, false);
        }
    }
    const float bb = bias ? bias[n0 + lr] : 0.f;
    const int rbase = m0 + 8 * half;
    #pragma unroll
    for (int r = 0; r < 8; ++r) {
        out[(size_t)(rbase      + r) * N + n0 + lr] = acc0[r] + bb;
        out[(size_t)(rbase + 16 + r) * N + n0 + lr] = acc1[r] + bb;
        out[(size_t)(rbase + 32 + r) * N + n0 + lr] = acc2[r] + bb;
        out[(size_t)(rbase + 48 + r) * N + n0 + lr] = acc3[r] + bb;
    }
}

// ---------------------------------------------------------------------------
// BatchNorm (training-mode batch stats over all rows), deterministic 2-stage.
// Stage 1: each block reduces 128 rows -> per-channel partial sum / sumsq.
// ---------------------------------------------------------------------------
__global__ void bn_stats(const float* __restrict__ x, float* __restrict__ part, int F)
{
    const int blk = blockIdx.x;
    const int r0 = blk * 128;
    for (int c = threadIdx.x; c < F; c += blockDim.x) {
        float s = 0.f, q = 0.f;
        for (int r = 0; r < 128; ++r) {
            const float v = x[(size_t)(r0 + r) * F + c];
            s += v; q += v * v;
        }
        part[(size_t)blk * 2 * F + c]     = s;
        part[(size_t)blk * 2 * F + F + c] = q;
    }
}

// Stage 2: fold partials, emit scale = g*rsqrt(var+eps), shift = beta - mean*scale
__global__ void bn_finalize(const float* __restrict__ part, int nblk, int F, float Minv,
                            const float* __restrict__ g, const float* __restrict__ beta,
                            float* __restrict__ ss)
{
    const int c = blockIdx.x * blockDim.x + threadIdx.x;
    if (c >= F) return;
    float s = 0.f, q = 0.f;
    for (int b = 0; b < nblk; ++b) {
        s += part[(size_t)b * 2 * F + c];
        q += part[(size_t)b * 2 * F + F + c];
    }
    const float mean = s * Minv;
    const float var  = q * Minv - mean * mean;
    const float inv  = rsqrtf(var + 1e-5f);
    const float sc   = g[c] * inv;
    ss[c]     = sc;
    ss[F + c] = beta[c] - mean * sc;
}

// y = relu(x*scale + shift) (+ residual)
__global__ void bn_apply(const float* __restrict__ x, const float* __restrict__ ss,
                         const float* __restrict__ res, float* __restrict__ out,
                         size_t total, int F)
{
    const size_t i = blockIdx.x * (size_t)blockDim.x + threadIdx.x;
    if (i >= total) return;
    const int c = (int)(i % F);
    float y = x[i] * ss[c] + ss[F + c];
    y = y > 0.f ? y : 0.f;
    if (res) y += res[i];
    out[i] = y;
}

// ---------------------------------------------------------------------------
// max_pool1d(kernel=4) over node dim with first-max argmax (jnp semantics).
// ---------------------------------------------------------------------------
__global__ void pool_kernel(const float* __restrict__ x, float* __restrict__ out,
                            int* __restrict__ argidx, int Vout, int F)
{
    const size_t i = blockIdx.x * (size_t)blockDim.x + threadIdx.x;
    const size_t total = (size_t)B_SZ * Vout * F;
    if (i >= total) return;
    const int f = (int)(i % F);
    const size_t t = i / F;
    const int p = (int)(t % Vout);
    const int b = (int)(t / Vout);
    const size_t base = ((size_t)b * (Vout * 4) + (size_t)p * 4) * F + f;
    float m = x[base];
    int arg = 0;
    #pragma unroll
    for (int j = 1; j < 4; ++j) {
        const float v = x[base + (size_t)j * F];
        if (v > m) { m = v; arg = j; }
    }
    out[i]    = m;
    argidx[i] = p * 4 + arg;
}

__global__ void zero_kernel(float* __restrict__ p, size_t n)
{
    const size_t i = blockIdx.x * (size_t)blockDim.x + threadIdx.x;
    if (i < n) p[i] = 0.f;
}

// scatter back: out[b, idx[b,p,f], f] = x[b,p,f]  (out pre-zeroed)
__global__ void unpool_kernel(const float* __restrict__ x, const int* __restrict__ argidx,
                              float* __restrict__ out, int Vin, int F, int Vout)
{
    const size_t i = blockIdx.x * (size_t)blockDim.x + threadIdx.x;
    const size_t total = (size_t)B_SZ * Vin * F;
    if (i >= total) return;
    const int f = (int)(i % F);
    const size_t t = i / F;
    const int b = (int)(t / Vin);
    out[((size_t)b * Vout + argidx[i]) * F + f] = x[i];
}

__global__ void concat_kernel(const float* __restrict__ a, int Fa,
                              const float* __restrict__ b, int Fb,
                              float* __restrict__ out, size_t rows)
{
    const size_t i = blockIdx.x * (size_t)blockDim.x + threadIdx.x;
    const int Fo = Fa + Fb;
    const size_t total = rows * Fo;
    if (i >= total) return;
    const int c = (int)(i % Fo);
    const size_t r = i / Fo;
    out[i] = (c < Fa) ? a[r * Fa + c] : b[r * Fb + (c - Fa)];
}

// ---------------------------------------------------------------------------
// Host orchestration
// ---------------------------------------------------------------------------
static inline unsigned ew(size_t n) { return (unsigned)((n + 255) / 256); }

static void do_lmul(const float* in, const int* li, const float* lv, const float* sub,
                    float* out, int V, int F, hipStream_t st)
{
    dim3 g(V, B_SZ);
    lmul_kernel<<<g, 64, 0, st>>>(in, li, lv, sub, out, V, F);
}

static void do_gemm(const float* A0, const float* A1, const float* A2, int nseg,
                    const float* W, const float* bias, float* out,
                    int M, int N, int F, hipStream_t st)
{
    dim3 g(M / 256, N / 16);          // 4 waves/block x 64 rows/wave
    gemm_wmma<<<g, 128, 0, st>>>(A0, A1, A2, W, bias, out, M, N, F, nseg);
}

static void do_bn(const float* xin, int M, int F, const float* g_, const float* beta,
                  const float* res, float* out, float* part, float* ss, hipStream_t st)
{
    const int nblk = M / 128;
    bn_stats<<<nblk, 256, 0, st>>>(xin, part, F);
    bn_finalize<<<(F + 63) / 64, 64, 0, st>>>(part, nblk, F, 1.0f / (float)M, g_, beta, ss);
    const size_t total = (size_t)M * F;
    bn_apply<<<ew(total), 256, 0, st>>>(xin, ss, res, out, total, F);
}

struct Blk { const float *W, *b, *beta, *g; };
struct Lin { const float *W, *b; };

extern "C" void kernel_launch(void* const* d_in, const int* in_sizes, int n_in,
                              void* d_out, int out_size, void* d_ws, size_t ws_size,
                              hipStream_t stream)
{
    const float* x   = (const float*)d_in[0];
    const int*   l0i = (const int*)  d_in[1];
    const float* l0v = (const float*)d_in[2];
    const int*   l1i = (const int*)  d_in[3];
    const float* l1v = (const float*)d_in[4];
    const int*   l2i = (const int*)  d_in[5];
    const float* l2v = (const float*)d_in[6];

    // Params flattened as a pytree: top-level keys sorted, then W,b,beta,g / W,b.
    int p = 7;
    auto getBlk = [&](Blk& o) {
        o.W = (const float*)d_in[p++]; o.b    = (const float*)d_in[p++];
        o.beta = (const float*)d_in[p++]; o.g = (const float*)d_in[p++];
    };
    auto getLin = [&](Lin& o) {
        o.W = (const float*)d_in[p++]; o.b = (const float*)d_in[p++];
    };
    Blk c11, c13, c21, c23, c31, c33, u11, u12, u21, u22;
    Lin r1, r2, r3, u13;
    getBlk(c11); getBlk(c13); getLin(r1);
    getBlk(c21); getBlk(c23); getLin(r2);
    getBlk(c31); getBlk(c33); getLin(r3);
    getBlk(u11); getBlk(u12); getLin(u13);
    getBlk(u21); getBlk(u22);
    if (p > n_in) return;

    // Workspace arena (floats)
    float* base = (float*)d_ws;
    size_t off = 0;
    auto alloc = [&](size_t n) { float* r = base + off; off += n; return r; };
    float* x11  = alloc((size_t)ROWS0 * 64);   // skip (conv11 out)
    float* x1s  = alloc((size_t)ROWS0 * 128);  // skip (conv13 + res)
    float* x2s  = alloc((size_t)ROWS1 * 256);  // skip (conv23 + res)
    int*   idx1 = (int*)alloc((size_t)ROWS1 * 128);
    int*   idx2 = (int*)alloc((size_t)ROWS2 * 256);
    float* t1   = alloc((size_t)ROWS0 * 256);  // Chebyshev T1 term
    float* t2   = alloc((size_t)ROWS0 * 256);  // Chebyshev T2 term
    float* bc   = alloc((size_t)ROWS0 * 256);  // concat / intermediate act
    float* b4   = alloc((size_t)ROWS0 * 128);  // gemm output
    float* b5   = alloc((size_t)ROWS0 * 128);  // residual / act
    float* b6   = alloc((size_t)ROWS0 * 128);  // pooled / act
    float* part = alloc(393216);               // bn partials
    float* ss   = alloc(1024);                 // bn scale/shift
    if (ws_size < off * sizeof(float)) return;

    float* out = (float*)d_out;
    hipStream_t st = stream;

    // ---- conv11: cheb(x, 64->64) @ L0, bn+relu -> x11 ----
    do_lmul(x, l0i, l0v, nullptr, t1, NV0, 64, st);
    do_lmul(t1, l0i, l0v, x, t2, NV0, 64, st);
    do_gemm(x, t1, t2, 3, c11.W, c11.b, b4, ROWS0, 64, 64, st);
    do_bn(b4, ROWS0, 64, c11.g, c11.beta, nullptr, x11, part, ss, st);

    // ---- conv13: cheb(x11, 64->128), bn+relu, + lin(x, 64->128) -> x1 ----
    do_lmul(x11, l0i, l0v, nullptr, t1, NV0, 64, st);
    do_lmul(t1, l0i, l0v, x11, t2, NV0, 64, st);
    do_gemm(x11, t1, t2, 3, c13.W, c13.b, b4, ROWS0, 128, 64, st);
    do_gemm(x, nullptr, nullptr, 1, r1.W, r1.b, b5, ROWS0, 128, 64, st);
    do_bn(b4, ROWS0, 128, c13.g, c13.beta, b5, x1s, part, ss, st);

    // ---- pool1: x1 -> x2i (b6), idx1 ----
    pool_kernel<<<ew((size_t)ROWS1 * 128), 256, 0, st>>>(x1s, b6, idx1, NV1, 128);

    // ---- conv21: cheb(x2i, 128->192) @ L1, bn+relu -> bc ----
    do_lmul(b6, l1i, l1v, nullptr, t1, NV1, 128, st);
    do_lmul(t1, l1i, l1v, b6, t2, NV1, 128, st);
    do_gemm(b6, t1, t2, 3, c21.W, c21.b, b4, ROWS1, 192, 128, st);
    do_bn(b4, ROWS1, 192, c21.g, c21.beta, nullptr, bc, part, ss, st);

    // ---- conv23: cheb(bc, 192->256), bn+relu, + lin(x2i, 128->256) -> x2 ----
    do_lmul(bc, l1i, l1v, nullptr, t1, NV1, 192, st);
    do_lmul(t1, l1i, l1v, bc, t2, NV1, 192, st);
    do_gemm(bc, t1, t2, 3, c23.W, c23.b, b4, ROWS1, 256, 192, st);
    do_gemm(b6, nullptr, nullptr, 1, r2.W, r2.b, b5, ROWS1, 256, 128, st);
    do_bn(b4, ROWS1, 256, c23.g, c23.beta, b5, x2s, part, ss, st);

    // ---- pool2: x2 -> x3i (b6), idx2 ----
    pool_kernel<<<ew((size_t)ROWS2 * 256), 256, 0, st>>>(x2s, b6, idx2, NV2, 256);

    // ---- conv31: cheb(x3i, 256->512) @ L2, bn+relu -> bc ----
    do_lmul(b6, l2i, l2v, nullptr, t1, NV2, 256, st);
    do_lmul(t1, l2i, l2v, b6, t2, NV2, 256, st);
    do_gemm(b6, t1, t2, 3, c31.W, c31.b, b4, ROWS2, 512, 256, st);
    do_bn(b4, ROWS2, 512, c31.g, c31.beta, nullptr, bc, part, ss, st);

    // ---- conv33: cheb(bc, 512->256), bn+relu, + lin(x3i, 256->256) -> x3 (b6) ----
    do_lmul(bc, l2i, l2v, nullptr, t1, NV2, 512, st);
    do_lmul(t1, l2i, l2v, bc, t2, NV2, 512, st);
    do_gemm(bc, t1, t2, 3, c33.W, c33.b, b4, ROWS2, 256, 512, st);
    do_gemm(b6, nullptr, nullptr, 1, r3.W, r3.b, b5, ROWS2, 256, 256, st);
    do_bn(b4, ROWS2, 256, c33.g, c33.beta, b5, b6, part, ss, st);   // b6 := x3

    // ---- unpool(x3, idx2) -> b4 [ROWS1,256]; concat with x2 -> bc [ROWS1,512] ----
    zero_kernel<<<ew((size_t)ROWS1 * 256), 256, 0, st>>>(b4, (size_t)ROWS1 * 256);
    unpool_kernel<<<ew((size_t)ROWS2 * 256), 256, 0, st>>>(b6, idx2, b4, NV2, 256, NV1);
    concat_kernel<<<ew((size_t)ROWS1 * 512), 256, 0, st>>>(b4, 256, x2s, 256, bc, (size_t)ROWS1);

    // ---- uconv21: cheb(bc, 512->256) @ L1, bn+relu -> b5 ----
    do_lmul(bc, l1i, l1v, nullptr, t1, NV1, 512, st);
    do_lmul(t1, l1i, l1v, bc, t2, NV1, 512, st);
    do_gemm(bc, t1, t2, 3, u21.W, u21.b, b4, ROWS1, 256, 512, st);
    do_bn(b4, ROWS1, 256, u21.g, u21.beta, nullptr, b5, part, ss, st);

    // ---- uconv22: cheb(b5, 256->128), bn+relu -> b6 ----
    do_lmul(b5, l1i, l1v, nullptr, t1, NV1, 256, st);
    do_lmul(t1, l1i, l1v, b5, t2, NV1, 256, st);
    do_gemm(b5, t1, t2, 3, u22.W, u22.b, b4, ROWS1, 128, 256, st);
    do_bn(b4, ROWS1, 128, u22.g, u22.beta, nullptr, b6, part, ss, st);

    // ---- unpool(b6, idx1) -> b4 [ROWS0,128]; concat with x1 -> bc [ROWS0,256] ----
    zero_kernel<<<ew((size_t)ROWS0 * 128), 256, 0, st>>>(b4, (size_t)ROWS0 * 128);
    unpool_kernel<<<ew((size_t)ROWS1 * 128), 256, 0, st>>>(b6, idx1, b4, NV1, 128, NV0);
    concat_kernel<<<ew((size_t)ROWS0 * 256), 256, 0, st>>>(b4, 128, x1s, 128, bc, (size_t)ROWS0);

    // ---- uconv11: cheb(bc, 256->128) @ L0, bn+relu -> b5 ----
    do_lmul(bc, l0i, l0v, nullptr, t1, NV0, 256, st);
    do_lmul(t1, l0i, l0v, bc, t2, NV0, 256, st);
    do_gemm(bc, t1, t2, 3, u11.W, u11.b, b4, ROWS0, 128, 256, st);
    do_bn(b4, ROWS0, 128, u11.g, u11.beta, nullptr, b5, part, ss, st);

    // ---- uconv12: cheb(b5, 128->64), bn+relu -> b6 ----
    do_lmul(b5, l0i, l0v, nullptr, t1, NV0, 128, st);
    do_lmul(t1, l0i, l0v, b5, t2, NV0, 128, st);
    do_gemm(b5, t1, t2, 3, u12.W, u12.b, b4, ROWS0, 64, 128, st);
    do_bn(b4, ROWS0, 64, u12.g, u12.beta, nullptr, b6, part, ss, st);

    // ---- final: cheb(concat(b6, x11) 128 -> 64) + bias, no bn -> d_out ----
    concat_kernel<<<ew((size_t)ROWS0 * 128), 256, 0, st>>>(b6, 64, x11, 64, bc, (size_t)ROWS0);
    do_lmul(bc, l0i, l0v, nullptr, t1, NV0, 128, st);
    do_lmul(t1, l0i, l0v, bc, t2, NV0, 128, st);
    do_gemm(bc, t1, t2, 3, u13.W, u13.b, out, ROWS0, 64, 128, st);
}